// DetectSpikes_48198122996104
// MI455X (gfx1250) — compile-verified
//
#include <hip/hip_runtime.h>
#include <stdint.h>

// ---------------- problem constants (from reference) ----------------
#define MCH      384           // channels
#define ADJW     12            // 384 / 32 mask words per row
#define TT       32            // time-tile size per block
#define HALO     15            // TIME_RADIUS
#define WIN      31            // 2*HALO+1
#define ROWS_L   (TT + 2*HALO) // 62 staged rows per tile
#define SEGS     (MCH / 4)     // 96 x 16B segments per row
#define MARGIN   20
#define NTHRESH  (-3.0f)       // raw-domain threshold (inv >= 3  <=>  raw <= -3)
#define RADIUS2  10000.0f      // 100^2
#define MAXDET   100000

typedef __attribute__((ext_vector_type(2))) float v2f;
typedef __attribute__((ext_vector_type(8))) float v8f;

__device__ __forceinline__ unsigned int ballot32(bool p) {
#if __has_builtin(__builtin_amdgcn_ballot_w32)
  return __builtin_amdgcn_ballot_w32(p);
#else
  return (unsigned int)(__ballot((int)p) & 0xffffffffull);
#endif
}

// CDNA5 async memory->LDS copy, 16B per lane, tracked by ASYNCcnt.
// GV mode: dst VGPR = LDS byte offset (generic LDS addr low 32 bits),
// addr VGPR pair = 64-bit global address.
__device__ __forceinline__ void async_copy16(void* ldsDst, const void* gSrc) {
#if defined(__AMDGCN__)
  unsigned int l = (unsigned int)(uintptr_t)ldsDst;
  unsigned long long g = (unsigned long long)(uintptr_t)gSrc;
  asm volatile("global_load_async_to_lds_b128 %0, %1, off"
               :: "v"(l), "v"(g) : "memory");
#else
  __builtin_memcpy(ldsDst, gSrc, 16);
#endif
}

__device__ __forceinline__ void async_wait0() {
#if defined(__AMDGCN__)
  asm volatile("s_wait_asynccnt 0x0" ::: "memory");
#endif
}

// -------- kernel 0: zero adjacency bitmask (atomicOr accumulator) --------
__global__ void k_zero(unsigned int* __restrict__ p, int n) {
  int i = blockIdx.x * blockDim.x + threadIdx.x;
  if (i < n) p[i] = 0u;
}

// -------- kernel 1: channel adjacency via WMMA Gram matrix --------
// G = X * X^T with X = [384 x 2] channel coords, K padded 2->4.
// dist^2(m,n) = |x_m|^2 + |x_n|^2 - 2*G[m][n]; adj bit set if <= R^2.
__global__ void __launch_bounds__(256) k_adj(const float* __restrict__ loc,
                                             unsigned int* __restrict__ adjw) {
  const int TILES = MCH / 16; // 24
  int wave = (blockIdx.x * blockDim.x + threadIdx.x) >> 5;
  int lane = threadIdx.x & 31;
  if (wave >= TILES * TILES) return;           // wave-uniform guard (EXEC stays full)
  int tm = (wave / TILES) * 16;
  int tn = (wave % TILES) * 16;
#if __has_builtin(__builtin_amdgcn_wmma_f32_16x16x4_f32)
  // A 16x4 f32 layout: lanes 0-15 -> K=0 (VGPR0), K=1 (VGPR1); lanes 16-31 -> K=2,3 (zero pad)
  v2f a = {0.0f, 0.0f};
  v2f b = {0.0f, 0.0f};
  if (lane < 16) {
    a.x = loc[2 * (tm + lane) + 0];
    a.y = loc[2 * (tm + lane) + 1];
    b.x = loc[2 * (tn + lane) + 0];
    b.y = loc[2 * (tn + lane) + 1];
  }
  v8f c = {};
  v8f d = __builtin_amdgcn_wmma_f32_16x16x4_f32(
      /*neg_a=*/false, a, /*neg_b=*/false, b,
      /*c_mod=*/(short)0, c, /*reuse_a=*/false, /*reuse_b=*/false);
  // C/D layout: VGPR v, lane L -> M = v + 8*(L>=16), N = L&15
  int nn = tn + (lane & 15);
  int mbase = tm + ((lane >> 4) << 3);
  float nx = loc[2 * nn], ny = loc[2 * nn + 1];
  float n2 = nx * nx + ny * ny;
#pragma unroll
  for (int v = 0; v < 8; ++v) {
    int mm = mbase + v;
    float mx = loc[2 * mm], my = loc[2 * mm + 1];
    float dist2 = (mx * mx + my * my) + n2 - 2.0f * d[v];
    if (dist2 <= RADIUS2)
      atomicOr(&adjw[mm * ADJW + (nn >> 5)], 1u << (nn & 31));
  }
#else
  int tid = blockIdx.x * blockDim.x + threadIdx.x;
  int nth = gridDim.x * blockDim.x;
  for (int p = tid; p < MCH * MCH; p += nth) {
    int m = p / MCH, n = p % MCH;
    float dx = loc[2 * m] - loc[2 * n];
    float dy = loc[2 * m + 1] - loc[2 * n + 1];
    if (dx * dx + dy * dy <= RADIUS2)
      atomicOr(&adjw[m * ADJW + (n >> 5)], 1u << (n & 31));
  }
#endif
}

// -------- kernel 2: fused temporal-min + spatial test -> detection bitmask --------
// Raw-domain formulation (no negation) so the tile can be staged verbatim with
// CDNA5 global_load_async_to_lds_b128. One block = one time tile x 384 channels,
// 384 threads (12 wave32). LDS: raw[62][384] + tmin[32][384] = 144,384 B dynamic.
__global__ void __launch_bounds__(MCH) k_detect(const float* __restrict__ traces,
                                                const unsigned int* __restrict__ adjw,
                                                unsigned int* __restrict__ mask,
                                                int N) {
  extern __shared__ float smem[];
  float* raw  = smem;                     // ROWS_L * MCH
  float* tmin = smem + ROWS_L * MCH;      // TT * MCH
  const int m = threadIdx.x;
  const long t0 = (long)blockIdx.x * TT;

  // stage raw tile (+halo) into LDS: async b128 for in-range rows,
  // +inf scalar fill for out-of-range halo rows (first/last blocks only)
  for (int s = threadIdx.x; s < ROWS_L * SEGS; s += MCH) {
    int row = s / SEGS;
    int c4  = (s - row * SEGS) * 4;
    long t  = t0 - HALO + row;
    float* dst = &raw[row * MCH + c4];
    if (t >= 0 && t < N) {
      async_copy16(dst, traces + t * (long)MCH + c4);
    } else {
      dst[0] = INFINITY; dst[1] = INFINITY; dst[2] = INFINITY; dst[3] = INFINITY;
    }
  }
  async_wait0();      // wave's own async copies complete (ASYNCcnt == 0)
  __syncthreads();    // cross-wave visibility

  // per-channel sliding temporal min (window 31) in raw domain
  for (int tl = 0; tl < TT; ++tl) {
    float mn = raw[tl * MCH + m];
#pragma unroll 6
    for (int k = 1; k < WIN; ++k) mn = fminf(mn, raw[(tl + k) * MCH + m]);
    tmin[tl * MCH + m] = mn;
  }
  __syncthreads();

  // preload this channel's adjacency bitmask
  unsigned int aw[ADJW];
#pragma unroll
  for (int w = 0; w < ADJW; ++w) aw[w] = adjw[m * ADJW + w];

  const int lane = m & 31;
  const int word = m >> 5;
  for (int tl = 0; tl < TT; ++tl) {
    long gt = t0 + tl;
    bool pred = false;
    if (gt < N) {
      float v = raw[(tl + HALO) * MCH + m];
      if (gt >= MARGIN && gt < (long)(N - MARGIN) && v <= NTHRESH) {
        bool ok = true;
        for (int w = 0; w < ADJW && ok; ++w) {
          unsigned int bits = aw[w];
          const float* trow = tmin + tl * MCH + w * 32;
          while (bits) {
            int b = __ffs(bits) - 1;
            bits &= bits - 1;
            if (trow[b] < v) { ok = false; break; }  // raw <= all neighbor window minima
          }
        }
        pred = ok;
      }
    }
    unsigned int ball = ballot32(pred);   // one word per wave == 32 channels
    if (gt < N && lane == 0) mask[gt * (long)ADJW + word] = ball;
  }
}

// -------- kernel 3: per-time-row detection counts --------
__global__ void k_rowcnt(const unsigned int* __restrict__ mask,
                         unsigned int* __restrict__ row_off, int N) {
  int r = blockIdx.x * blockDim.x + threadIdx.x;
  if (r >= N) return;
  unsigned int c = 0;
#pragma unroll
  for (int w = 0; w < ADJW; ++w) c += __popc(mask[(long)r * ADJW + w]);
  row_off[r] = c;
}

// -------- kernel 4: single-block exclusive scan over row counts (in place) --------
__global__ void __launch_bounds__(1024) k_scan(unsigned int* __restrict__ row_off,
                                               unsigned int* __restrict__ total, int N) {
  __shared__ unsigned int s[1024];
  const int tid = threadIdx.x;
  unsigned int base = 0;
  int chunks = (N + 1023) / 1024;
  for (int c = 0; c < chunks; ++c) {
    int r = c * 1024 + tid;
    unsigned int v = (r < N) ? row_off[r] : 0u;
    s[tid] = v;
    __syncthreads();
    for (int off = 1; off < 1024; off <<= 1) {  // Hillis-Steele inclusive scan
      unsigned int t = (tid >= off) ? s[tid - off] : 0u;
      __syncthreads();
      s[tid] += t;
      __syncthreads();
    }
    if (r < N) row_off[r] = base + s[tid] - v;  // exclusive + running base
    unsigned int chunk_total = s[1023];
    __syncthreads();
    base += chunk_total;
  }
  if (tid == 0) *total = base;
}

// -------- kernel 5: fill outputs with -1 padding --------
__global__ void k_fill(long long* __restrict__ times, int* __restrict__ chans) {
  int i = blockIdx.x * blockDim.x + threadIdx.x;
  if (i < MAXDET) { times[i] = -1ll; chans[i] = -1; }
}

// -------- kernel 6: ordered emission (row-major == stable sort by time) --------
__global__ void k_emit(const unsigned int* __restrict__ mask,
                       const unsigned int* __restrict__ row_off,
                       long long* __restrict__ times, int* __restrict__ chans, int N) {
  int r = blockIdx.x * blockDim.x + threadIdx.x;
  if (r >= N) return;
  unsigned int idx = row_off[r];
  for (int w = 0; w < ADJW; ++w) {
    unsigned int bits = mask[(long)r * ADJW + w];
    while (bits) {
      int b = __ffs(bits) - 1;
      bits &= bits - 1;
      if (idx < MAXDET) { times[idx] = (long long)r; chans[idx] = w * 32 + b; }
      ++idx;
    }
  }
}

extern "C" void kernel_launch(void* const* d_in, const int* in_sizes, int n_in,
                              void* d_out, int out_size, void* d_ws, size_t ws_size,
                              hipStream_t stream) {
  const float* traces = (const float*)d_in[0];        // [N, 384] f32
  const float* loc    = (const float*)d_in[1];        // [384, 2] f32
  const int N = in_sizes[0] / MCH;

  // workspace layout
  char* ws = (char*)d_ws;
  unsigned int* adjw = (unsigned int*)ws;                           // 384*12*4 = 18,432 B
  unsigned int* mask = (unsigned int*)(ws + 32768);                 // N*12*4   = 7.2 MB
  size_t maskBytes = ((size_t)N * ADJW * 4 + 255) & ~(size_t)255;
  unsigned int* row_off = (unsigned int*)(ws + 32768 + maskBytes);  // N*4
  size_t rowBytes = ((size_t)N * 4 + 255) & ~(size_t)255;
  unsigned int* total = (unsigned int*)(ws + 32768 + maskBytes + rowBytes);

  long long* times = (long long*)d_out;          // int64 times, then int32 chans
  int* chans = (int*)(times + MAXDET);

  // 1) zero adjacency accumulator
  k_zero<<<(MCH * ADJW + 255) / 256, 256, 0, stream>>>(adjw, MCH * ADJW);
  // 2) adjacency via WMMA Gram matrix: 576 tiles, 8 waves/block -> 72 blocks
  k_adj<<<(MCH / 16) * (MCH / 16) / 8, 256, 0, stream>>>(loc, adjw);
  // 3) fused temporal-min + spatial detection (async-to-LDS staged)
  int nblk = (N + TT - 1) / TT;
  size_t smem = (size_t)(ROWS_L + TT) * MCH * sizeof(float);  // 144,384 B
  k_detect<<<nblk, MCH, smem, stream>>>(traces, adjw, mask, N);
  // 4) ordered compaction
  k_rowcnt<<<(N + 255) / 256, 256, 0, stream>>>(mask, row_off, N);
  k_scan<<<1, 1024, 0, stream>>>(row_off, total, N);
  k_fill<<<(MAXDET + 255) / 256, 256, 0, stream>>>(times, chans);
  k_emit<<<(N + 255) / 256, 256, 0, stream>>>(mask, row_off, times, chans, N);
}